// CenterLoss_40621800685615
// MI455X (gfx1250) — compile-verified
//
#include <hip/hip_runtime.h>
#include <hip/hip_bf16.h>

// ---------------------------------------------------------------------------
// CenterLoss forward for MI455X (gfx1250, wave32, WMMA).
//   out[0]         = mean(clamp(||x_i - centers[labels_i]||^2, 1e-12, 1e12))
//   out[1 + b*C+c] = (PReLU(x) @ fc3_w^T + fc3_b)[b][c]
//
// GEMM: V_WMMA_F32_16X16X4_F32 (full f32 on the matrix pipe).
//  - b128 operand loads: one 4-float quad per lane feeds TWO WMMA K-steps
//    (WMMA K-slot permutation is legal as long as A and B agree).
//  - software pipelined: next iteration's 6 loads are issued before the
//    current iteration's 16 WMMAs, hiding L2 latency.
//  - logits (819 MB) stored non-temporal so the L2 keeps fc3_w/x resident.
// ---------------------------------------------------------------------------

typedef __attribute__((ext_vector_type(2))) float v2f;
typedef __attribute__((ext_vector_type(4))) float v4f;
typedef __attribute__((ext_vector_type(8))) float v8f;

#define BQ 2048
#define CQ 100000
#define DQ 128

// ---------------------------------------------------------------------------
__global__ void zero_loss_kernel(float* __restrict__ out) {
  if (threadIdx.x == 0 && blockIdx.x == 0) out[0] = 0.0f;
}

// ---------------------------------------------------------------------------
// Center-loss branch: one block of 128 threads per sample.
__global__ __launch_bounds__(128) void center_loss_kernel(
    const float* __restrict__ x, const float* __restrict__ centers,
    const int* __restrict__ labels, float* __restrict__ loss_out) {
  const int i = blockIdx.x;
  const int d = threadIdx.x;
  const int lab = labels[i];

  const float diff = x[(size_t)i * DQ + d] - centers[(size_t)lab * DQ + d];
  float sq = diff * diff;

  #pragma unroll
  for (int off = 16; off > 0; off >>= 1) sq += __shfl_down(sq, off, 32);

  __shared__ float red[4];
  if ((threadIdx.x & 31) == 0) red[threadIdx.x >> 5] = sq;
  __syncthreads();

  if (threadIdx.x == 0) {
    float s = red[0] + red[1] + red[2] + red[3];
    s = fminf(fmaxf(s, 1e-12f), 1e12f);
    atomicAdd(loss_out, s * (1.0f / (float)BQ));
  }
}

// ---------------------------------------------------------------------------
__device__ __forceinline__ v4f prelu4(v4f v, float s) {
  v4f r;
  r.x = fmaxf(v.x, 0.0f) + s * fminf(v.x, 0.0f);
  r.y = fmaxf(v.y, 0.0f) + s * fminf(v.y, 0.0f);
  r.z = fmaxf(v.z, 0.0f) + s * fminf(v.z, 0.0f);
  r.w = fmaxf(v.w, 0.0f) + s * fminf(v.w, 0.0f);
  return r;
}

__device__ __forceinline__ v8f wmma_f32(v2f a, v2f b, v8f c) {
  return __builtin_amdgcn_wmma_f32_16x16x4_f32(false, a, false, b, (short)0,
                                               c, false, false);
}

// ---------------------------------------------------------------------------
// PReLU + GEMM + bias.
// Block = 256 threads = 8 waves (2 M x 4 N). Wave tile = 32(M) x 64(N).
// Block tile = 64 x 256. Grid = (ceil(C/256), B/64) = (391, 32).
// K loop: 16 iterations x 8 k's, 6 b128 loads + 16 WMMAs per iteration.
__global__ __launch_bounds__(256) void prelu_gemm_kernel(
    const float* __restrict__ x,        // [B, D]
    const float* __restrict__ prelu_a,  // [1]
    const float* __restrict__ w,        // [C, D]
    const float* __restrict__ bias,     // [C]
    float* __restrict__ out)            // logits base (d_out + 1)
{
  const int lane = threadIdx.x & 31;
  const int wave = threadIdx.x >> 5;
  const int wm = wave >> 2;   // 0..1
  const int wn = wave & 3;    // 0..3
  const int half = lane >> 4; // selects K quad (loads) / M+8 (stores)
  const int l15 = lane & 15;

  const int mBase = blockIdx.y * 64 + wm * 32;
  const int nBase = blockIdx.x * 256 + wn * 64;

  const float slope = prelu_a[0];

  v8f acc[2][4] = {};

  // Per-lane base pointers; K advances by immediate offsets (i*8 floats).
  const float* aP0 = x + (size_t)(mBase + l15) * DQ + half * 4;
  const float* aP1 = x + (size_t)(mBase + 16 + l15) * DQ + half * 4;
  const float* bP[4];
  #pragma unroll
  for (int nt = 0; nt < 4; ++nt) {
    int n = nBase + nt * 16 + l15;
    n = (n < CQ) ? n : (CQ - 1);  // clamp loads; stores are predicated
    bP[nt] = w + (size_t)n * DQ + half * 4;
  }

  // Prologue: iteration 0 quads.
  v4f a0q = *(const v4f*)aP0;
  v4f a1q = *(const v4f*)aP1;
  v4f bq0 = *(const v4f*)bP[0];
  v4f bq1 = *(const v4f*)bP[1];
  v4f bq2 = *(const v4f*)bP[2];
  v4f bq3 = *(const v4f*)bP[3];

  #pragma unroll
  for (int i = 0; i < DQ / 8; ++i) {
    // ---- issue next iteration's loads first (hide L2 latency) ----
    v4f a0n, a1n, bn0, bn1, bn2, bn3;
    if (i + 1 < DQ / 8) {
      const int off = (i + 1) * 8;
      a0n = *(const v4f*)(aP0 + off);
      a1n = *(const v4f*)(aP1 + off);
      bn0 = *(const v4f*)(bP[0] + off);
      bn1 = *(const v4f*)(bP[1] + off);
      bn2 = *(const v4f*)(bP[2] + off);
      bn3 = *(const v4f*)(bP[3] + off);
    }

    // ---- PReLU on current A quads ----
    const v4f h0 = prelu4(a0q, slope);
    const v4f h1 = prelu4(a1q, slope);

    // Step X: quad elements (0,1) -> global k {k0,k0+1,k0+4,k0+5}
    // Step Y: quad elements (2,3) -> global k {k0+2,k0+3,k0+6,k0+7}
    const v2f a0x = {h0.x, h0.y}, a0y = {h0.z, h0.w};
    const v2f a1x = {h1.x, h1.y}, a1y = {h1.z, h1.w};
    const v2f b0x = {bq0.x, bq0.y}, b0y = {bq0.z, bq0.w};
    const v2f b1x = {bq1.x, bq1.y}, b1y = {bq1.z, bq1.w};
    const v2f b2x = {bq2.x, bq2.y}, b2y = {bq2.z, bq2.w};
    const v2f b3x = {bq3.x, bq3.y}, b3y = {bq3.z, bq3.w};

    // 16 WMMAs: 8 independent accumulator chains, 2 steps each.
    acc[0][0] = wmma_f32(a0x, b0x, acc[0][0]);
    acc[1][0] = wmma_f32(a1x, b0x, acc[1][0]);
    acc[0][1] = wmma_f32(a0x, b1x, acc[0][1]);
    acc[1][1] = wmma_f32(a1x, b1x, acc[1][1]);
    acc[0][2] = wmma_f32(a0x, b2x, acc[0][2]);
    acc[1][2] = wmma_f32(a1x, b2x, acc[1][2]);
    acc[0][3] = wmma_f32(a0x, b3x, acc[0][3]);
    acc[1][3] = wmma_f32(a1x, b3x, acc[1][3]);
    acc[0][0] = wmma_f32(a0y, b0y, acc[0][0]);
    acc[1][0] = wmma_f32(a1y, b0y, acc[1][0]);
    acc[0][1] = wmma_f32(a0y, b1y, acc[0][1]);
    acc[1][1] = wmma_f32(a1y, b1y, acc[1][1]);
    acc[0][2] = wmma_f32(a0y, b2y, acc[0][2]);
    acc[1][2] = wmma_f32(a1y, b2y, acc[1][2]);
    acc[0][3] = wmma_f32(a0y, b3y, acc[0][3]);
    acc[1][3] = wmma_f32(a1y, b3y, acc[1][3]);

    // ---- rotate pipeline registers ----
    if (i + 1 < DQ / 8) {
      a0q = a0n; a1q = a1n;
      bq0 = bn0; bq1 = bn1; bq2 = bn2; bq3 = bn3;
    }
  }

  // Epilogue: bias + non-temporal streaming stores (keep L2 for W/x).
  // C/D layout: VGPR r -> M=r (lanes 0-15) / M=r+8 (lanes 16-31), N=l15.
  #pragma unroll
  for (int nt = 0; nt < 4; ++nt) {
    const int n = nBase + nt * 16 + l15;
    if (n < CQ) {
      const float bv = bias[n];
      #pragma unroll
      for (int r = 0; r < 8; ++r) {
        const int m0 = mBase + r + half * 8;       // first 16-row tile
        const int m1 = mBase + 16 + r + half * 8;  // second 16-row tile
        __builtin_nontemporal_store(acc[0][nt][r] + bv,
                                    &out[(size_t)m0 * CQ + n]);
        __builtin_nontemporal_store(acc[1][nt][r] + bv,
                                    &out[(size_t)m1 * CQ + n]);
      }
    }
  }
}

// ---------------------------------------------------------------------------
extern "C" void kernel_launch(void* const* d_in, const int* in_sizes, int n_in,
                              void* d_out, int out_size, void* d_ws, size_t ws_size,
                              hipStream_t stream) {
  (void)in_sizes; (void)n_in; (void)out_size; (void)d_ws; (void)ws_size;

  const float* x       = (const float*)d_in[0];
  const float* centers = (const float*)d_in[1];
  const float* prelu_a = (const float*)d_in[2];
  const float* fc3_w   = (const float*)d_in[3];
  const float* fc3_b   = (const float*)d_in[4];
  const int*   labels  = (const int*)d_in[5];

  float* out = (float*)d_out;

  zero_loss_kernel<<<1, 32, 0, stream>>>(out);
  center_loss_kernel<<<BQ, 128, 0, stream>>>(x, centers, labels, out);

  dim3 grid((CQ + 255) / 256, BQ / 64);
  prelu_gemm_kernel<<<grid, 256, 0, stream>>>(x, prelu_a, fc3_w, fc3_b, out + 1);
}